// Alpha2Assoc_68667937129117
// MI455X (gfx1250) — compile-verified
//
#include <hip/hip_runtime.h>

// Alpha-to-associativity visibility kernel for MI455X (gfx1250, wave32).
//
// Pure streaming scan: 128 MiB in + 384 MiB out => bandwidth-bound
// (~22 us at 23.3 TB/s roofline). No matrix contraction exists in this op
// (arithmetic intensity ~0.75 flop/byte), so WMMA is inapplicable; the
// CDNA5-relevant levers are: single-pass fusion of the 3 layers (only the
// three running transmittances p0,p1,p2 stay live), 128-bit coalesced VMEM,
// non-temporal cache hints (every byte touched exactly once), speculative
// global_prefetch_b8 of the next d-plane, and wave32-sized 256-thread blocks.

typedef __attribute__((ext_vector_type(4))) float v4f;

namespace {
constexpr int   kD        = 32;          // planes
constexpr int   kHW       = 512 * 512;   // pixels per (b,d) plane = 2^18
constexpr int   kVec      = 4;           // pixels per thread (float4)
constexpr float kAlphaMin = 1e-4f;

__device__ inline v4f vmaxs(v4f a, float s) {
    v4f r;
    r.x = fmaxf(a.x, s);
    r.y = fmaxf(a.y, s);
    r.z = fmaxf(a.z, s);
    r.w = fmaxf(a.w, s);
    return r;
}
} // namespace

__global__ __launch_bounds__(256) void alpha2assoc_fused_kernel(
        const float* __restrict__ in, float* __restrict__ out, int n_groups) {
    const int g = blockIdx.x * blockDim.x + threadIdx.x;  // group of 4 pixels
    if (g >= n_groups) return;

    const long q   = (long)g * kVec;          // flat (b, pixel) index
    const int  b   = (int)(q >> 18);          // q / kHW  (kHW = 2^18)
    const int  pix = (int)(q & (kHW - 1));    // q % kHW  (16B aligned: q mult. of 4)

    const float* ip = in  + (size_t)b * kD * kHW + pix;          // stride kHW per d
    float*       op = out + (size_t)b * kD * 3 * kHW + pix;      // stride 3*kHW per d

    // Running transmittance (exclusive prefix product of (1 - a_L)) per layer.
    v4f p0 = {1.f, 1.f, 1.f, 1.f};
    v4f p1 = {1.f, 1.f, 1.f, 1.f};
    v4f p2 = {1.f, 1.f, 1.f, 1.f};

#pragma unroll 8
    for (int d = 0; d < kD; ++d) {
        // Speculative prefetch of the next d-plane (global_prefetch_b8, TH=0).
        // Unconditional: ISA 10.5 — speculative prefetch past the allocation is
        // silently dropped on translation failure, no error, no counter.
        __builtin_prefetch(ip + kHW, 0, 0);

        // Streamed once: non-temporal 128-bit load.
        v4f a = __builtin_nontemporal_load((const v4f*)ip);
        a = vmaxs(a, kAlphaMin);                       // clip(alpha, 1e-4)

        // ---- layer 0 (incoming occ == 1) ----
        const v4f vis0 = p0;                           // exclusive cumprod
        const v4f occ1 = 1.0f - vis0;                  // occlusion into layer 1
        __builtin_nontemporal_store(vis0, (v4f*)op);   // vis0 * occ0(=1)
        p0 = p0 * (1.0f - a);

        // ---- layer 1 ----
        const v4f a1   = a * occ1;                     // attenuated alpha
        const v4f vis1 = p1;
        const v4f occ2 = 1.0f - vis1;
        __builtin_nontemporal_store(vis1 * occ1, (v4f*)(op + kHW));
        p1 = p1 * (1.0f - a1);

        // ---- layer 2 ----
        const v4f a2   = a1 * occ2;
        const v4f vis2 = p2;
        __builtin_nontemporal_store(vis2 * occ2, (v4f*)(op + 2 * kHW));
        p2 = p2 * (1.0f - a2);

        ip += kHW;
        op += 3 * kHW;
    }
}

extern "C" void kernel_launch(void* const* d_in, const int* in_sizes, int n_in,
                              void* d_out, int out_size, void* d_ws, size_t ws_size,
                              hipStream_t stream) {
    (void)n_in; (void)out_size; (void)d_ws; (void)ws_size;

    const float* in  = (const float*)d_in[0];
    float*       out = (float*)d_out;

    // in_sizes[0] = B * D * H * W ; pixels = that / D ; 4 pixels per thread.
    const long total    = (long)in_sizes[0];
    const int  n_groups = (int)(total / kD / kVec);   // 262,144 for the reference shape

    const int threads = 256;                           // 8 wave32 waves per block
    const int blocks  = (n_groups + threads - 1) / threads;

    alpha2assoc_fused_kernel<<<blocks, threads, 0, stream>>>(in, out, n_groups);
}